// PortfolioRiskManager_58042188038540
// MI455X (gfx1250) — compile-verified
//
#include <hip/hip_runtime.h>

#define THREADS     256
#define NBLK        1024
#define NUM_SECTORS 11
#define VOL_SCALAR  0.75f     /* 0.15 / 0.2 exactly */
#define MAX_IND     0.01f
#define MAX_SEC     0.15f

// ---------------------------------------------------------------------------
// Pass A: per-block sector abs-sums of clip(p*VS) and per-block cap sum.
// partials layout: [block][12] = sector 0..10 abs-sum, [11] = cap sum.
// ---------------------------------------------------------------------------
__global__ __launch_bounds__(THREADS) void pr_passA(
    const float* __restrict__ pos,
    const int*   __restrict__ sid,
    const float* __restrict__ mc,
    long long n,
    float* __restrict__ partials)
{
    __shared__ float red[THREADS];
    const int t = threadIdx.x;

    const long long nvec   = n >> 2;
    const long long stride = (long long)gridDim.x * THREADS;
    const float4* __restrict__ p4 = reinterpret_cast<const float4*>(pos);
    const int4*   __restrict__ s4 = reinterpret_cast<const int4*>(sid);
    const float4* __restrict__ m4 = reinterpret_cast<const float4*>(mc);

    float bins[NUM_SECTORS];
#pragma unroll
    for (int s = 0; s < NUM_SECTORS; ++s) bins[s] = 0.0f;
    float capsum = 0.0f;

    for (long long i = (long long)blockIdx.x * THREADS + t; i < nvec; i += stride) {
        const long long ip = i + stride;
        if (ip < nvec) {            // gfx1250 global_prefetch_b8 path
            __builtin_prefetch(p4 + ip, 0, 1);
            __builtin_prefetch(s4 + ip, 0, 1);
            __builtin_prefetch(m4 + ip, 0, 1);
        }
        const float4 p = p4[i];
        const int4   s = s4[i];
        const float4 m = m4[i];
        capsum += (m.x + m.y) + (m.z + m.w);
        // |clip(x*VS, +-c)| == min(|x|*VS, c)
        const float a0 = fminf(fabsf(p.x) * VOL_SCALAR, MAX_IND);
        const float a1 = fminf(fabsf(p.y) * VOL_SCALAR, MAX_IND);
        const float a2 = fminf(fabsf(p.z) * VOL_SCALAR, MAX_IND);
        const float a3 = fminf(fabsf(p.w) * VOL_SCALAR, MAX_IND);
#pragma unroll
        for (int k = 0; k < NUM_SECTORS; ++k) {
            float acc = 0.0f;
            acc += (s.x == k) ? a0 : 0.0f;
            acc += (s.y == k) ? a1 : 0.0f;
            acc += (s.z == k) ? a2 : 0.0f;
            acc += (s.w == k) ? a3 : 0.0f;
            bins[k] += acc;
        }
    }
    // scalar tail (n % 4)
    for (long long i = nvec * 4 + (long long)blockIdx.x * THREADS + t; i < n; i += stride) {
        const float a = fminf(fabsf(pos[i]) * VOL_SCALAR, MAX_IND);
        const int   s = sid[i];
        capsum += mc[i];
#pragma unroll
        for (int k = 0; k < NUM_SECTORS; ++k) bins[k] += (s == k) ? a : 0.0f;
    }

    // 12 deterministic block tree-reductions
#pragma unroll 1
    for (int c = 0; c < NUM_SECTORS + 1; ++c) {
        red[t] = (c < NUM_SECTORS) ? bins[c] : capsum;
        __syncthreads();
        for (int off = THREADS / 2; off > 0; off >>= 1) {
            if (t < off) red[t] += red[t + off];
            __syncthreads();
        }
        if (t == 0) partials[(long long)blockIdx.x * 12 + c] = red[0];
        __syncthreads();
    }
}

// ---------------------------------------------------------------------------
// Pass B: reduce block partials -> scalars[0..10]=sector scale, [11]=1/capsum
// ---------------------------------------------------------------------------
__global__ __launch_bounds__(THREADS) void pr_passB(
    const float* __restrict__ partials, float* __restrict__ scalars)
{
    __shared__ float red[THREADS];
    const int t = threadIdx.x;
#pragma unroll 1
    for (int c = 0; c < NUM_SECTORS + 1; ++c) {
        float s = 0.0f;
        for (int b = t; b < NBLK; b += THREADS) s += partials[b * 12 + c];
        red[t] = s;
        __syncthreads();
        for (int off = THREADS / 2; off > 0; off >>= 1) {
            if (t < off) red[t] += red[t + off];
            __syncthreads();
        }
        if (t == 0) {
            const float v = red[0];
            if (c < NUM_SECTORS) scalars[c] = (v > MAX_SEC) ? (MAX_SEC / v) : 1.0f;
            else                 scalars[NUM_SECTORS] = 1.0f / v;  // inv cap sum
        }
        __syncthreads();
    }
}

// ---------------------------------------------------------------------------
// Pass C: out = clip(p*VS)*scale[sid]*sqrt(mc/capsum); per-block sum |out|
// ---------------------------------------------------------------------------
__global__ __launch_bounds__(THREADS) void pr_passC(
    const float* __restrict__ pos,
    const int*   __restrict__ sid,
    const float* __restrict__ mc,
    const float* __restrict__ scalars,
    long long n,
    float* __restrict__ out,
    float* __restrict__ partialC)
{
    __shared__ float sscl[NUM_SECTORS + 1];
    __shared__ float red[THREADS];
    const int t = threadIdx.x;
    if (t < NUM_SECTORS + 1) sscl[t] = scalars[t];
    __syncthreads();
    const float inv_capsum = sscl[NUM_SECTORS];

    const long long nvec   = n >> 2;
    const long long stride = (long long)gridDim.x * THREADS;
    const float4* __restrict__ p4 = reinterpret_cast<const float4*>(pos);
    const int4*   __restrict__ s4 = reinterpret_cast<const int4*>(sid);
    const float4* __restrict__ m4 = reinterpret_cast<const float4*>(mc);
    float4* __restrict__       o4 = reinterpret_cast<float4*>(out);

    float acc = 0.0f;
    for (long long i = (long long)blockIdx.x * THREADS + t; i < nvec; i += stride) {
        const long long ip = i + stride;
        if (ip < nvec) {
            __builtin_prefetch(p4 + ip, 0, 1);
            __builtin_prefetch(s4 + ip, 0, 1);
            __builtin_prefetch(m4 + ip, 0, 1);
        }
        const float4 p = p4[i];
        const int4   s = s4[i];
        const float4 m = m4[i];
        const float v0 = fminf(fmaxf(p.x * VOL_SCALAR, -MAX_IND), MAX_IND);
        const float v1 = fminf(fmaxf(p.y * VOL_SCALAR, -MAX_IND), MAX_IND);
        const float v2 = fminf(fmaxf(p.z * VOL_SCALAR, -MAX_IND), MAX_IND);
        const float v3 = fminf(fmaxf(p.w * VOL_SCALAR, -MAX_IND), MAX_IND);
        float4 o;
        o.x = v0 * sscl[s.x] * sqrtf(m.x * inv_capsum);
        o.y = v1 * sscl[s.y] * sqrtf(m.y * inv_capsum);
        o.z = v2 * sscl[s.z] * sqrtf(m.z * inv_capsum);
        o.w = v3 * sscl[s.w] * sqrtf(m.w * inv_capsum);
        acc += (fabsf(o.x) + fabsf(o.y)) + (fabsf(o.z) + fabsf(o.w));
        o4[i] = o;
    }
    for (long long i = nvec * 4 + (long long)blockIdx.x * THREADS + t; i < n; i += stride) {
        const float v = fminf(fmaxf(pos[i] * VOL_SCALAR, -MAX_IND), MAX_IND);
        const float r = v * sscl[sid[i]] * sqrtf(mc[i] * inv_capsum);
        out[i] = r;
        acc += fabsf(r);
    }

    red[t] = acc;
    __syncthreads();
    for (int off = THREADS / 2; off > 0; off >>= 1) {
        if (t < off) red[t] += red[t + off];
        __syncthreads();
    }
    if (t == 0) partialC[blockIdx.x] = red[0];
}

// ---------------------------------------------------------------------------
// Pass D: total gross exposure -> scalars[12] = (total > 1) ? 1/total : 1
// ---------------------------------------------------------------------------
__global__ __launch_bounds__(THREADS) void pr_passD(
    const float* __restrict__ partialC, float* __restrict__ scalars)
{
    __shared__ float red[THREADS];
    const int t = threadIdx.x;
    float s = 0.0f;
    for (int b = t; b < NBLK; b += THREADS) s += partialC[b];
    red[t] = s;
    __syncthreads();
    for (int off = THREADS / 2; off > 0; off >>= 1) {
        if (t < off) red[t] += red[t + off];
        __syncthreads();
    }
    if (t == 0) {
        const float total = red[0];
        scalars[12] = (total > 1.0f) ? (1.0f / total) : 1.0f;
    }
}

// ---------------------------------------------------------------------------
// Pass E: out *= factor.
// If factor == 1.0f (the overwhelmingly common case: expected gross exposure
// after sector scaling + cap weighting is ~1e-3 << 1), multiplying by 1.0f is
// the exact identity on every float, so the whole pass is a no-op and each
// block early-exits after one scalar load — saving 8 B/elem of HBM traffic.
// Deterministic: same inputs -> same factor -> same work.
// ---------------------------------------------------------------------------
__global__ __launch_bounds__(THREADS) void pr_passE(
    float* __restrict__ out, const float* __restrict__ scalars, long long n)
{
    const float f = scalars[12];
    if (f == 1.0f) return;

    const long long nvec   = n >> 2;
    const long long stride = (long long)gridDim.x * THREADS;
    float4* __restrict__ o4 = reinterpret_cast<float4*>(out);
    for (long long i = (long long)blockIdx.x * THREADS + threadIdx.x; i < nvec; i += stride) {
        float4 v = o4[i];
        v.x *= f; v.y *= f; v.z *= f; v.w *= f;
        o4[i] = v;
    }
    for (long long i = nvec * 4 + (long long)blockIdx.x * THREADS + threadIdx.x; i < n; i += stride)
        out[i] *= f;
}

// ---------------------------------------------------------------------------
// gfx1250 async-to-LDS path (ASYNCcnt). Compiles and lowers on this toolchain
// (4-arg builtin, AS1/AS3 int* params). Kept as a compiled demonstration of
// the CDNA5 async data-movement path; the streaming passes themselves are
// HBM-bound with float4 loads, so LDS staging adds no bandwidth.
// ---------------------------------------------------------------------------
#if defined(__has_builtin)
#if __has_builtin(__builtin_amdgcn_global_load_async_to_lds_b32)
#define PR_HAVE_ASYNC_LDS 1
#endif
#endif

#ifdef PR_HAVE_ASYNC_LDS
typedef __attribute__((address_space(1))) int as1_i;
typedef __attribute__((address_space(3))) int as3_i;
__global__ __launch_bounds__(THREADS) void pr_async_probe(
    const float* __restrict__ g, float* __restrict__ o)
{
    __shared__ float tile[THREADS];
    const int t = threadIdx.x;
    const float* gp = g + (long long)blockIdx.x * THREADS + t;
    __builtin_amdgcn_global_load_async_to_lds_b32((as1_i*)gp, (as3_i*)&tile[t], 0, 0);
#if __has_builtin(__builtin_amdgcn_s_wait_asynccnt)
    __builtin_amdgcn_s_wait_asynccnt(0);
#else
    asm volatile("s_wait_asynccnt 0" ::: "memory");
#endif
    __syncthreads();
    o[(long long)blockIdx.x * THREADS + t] = tile[t];
}
#endif

// ---------------------------------------------------------------------------
extern "C" void kernel_launch(void* const* d_in, const int* in_sizes, int n_in,
                              void* d_out, int out_size, void* d_ws, size_t ws_size,
                              hipStream_t stream)
{
    const float* pos = (const float*)d_in[0];
    // d_in[1] = asset_ids: unused by the reference math
    const int*   sid = (const int*)d_in[2];
    const float* mc  = (const float*)d_in[3];
    float*       out = (float*)d_out;
    const long long n = (long long)in_sizes[0];

    float* partials = (float*)d_ws;                  // NBLK * 12
    float* scalars  = partials + (size_t)NBLK * 12;  // 16 (0..10 scale, 11 inv_capsum, 12 factor)
    float* partialC = scalars + 16;                  // NBLK

    pr_passA<<<NBLK, THREADS, 0, stream>>>(pos, sid, mc, n, partials);
    pr_passB<<<1,    THREADS, 0, stream>>>(partials, scalars);
    pr_passC<<<NBLK, THREADS, 0, stream>>>(pos, sid, mc, scalars, n, out, partialC);
    pr_passD<<<1,    THREADS, 0, stream>>>(partialC, scalars);
    pr_passE<<<NBLK, THREADS, 0, stream>>>(out, scalars, n);
}